// ContextAwareAttention_50637664420273
// MI455X (gfx1250) — compile-verified
//
#include <hip/hip_runtime.h>
#include <math.h>

typedef __attribute__((ext_vector_type(2))) float v2f;
typedef __attribute__((ext_vector_type(4))) float f4;
typedef __attribute__((ext_vector_type(8))) float v8f;

#define N_PROP 2048
#define B_SC   32
#define C_IN   256
#define C_P    32
#define HW     196   // 14*14 proposal spatial
#define HWP    208   // padded to 13 tiles of 16
#define SH     56
#define SHW    3136  // 56*56
#define PP     196   // POOL*POOL

// ---------------------------------------------------------------------------
// Kernel A: q[b,c] = sum_{y,x} scene[b,c,y,x] * w_fc[(y>>2)*14+(x>>2)] / 16
// One wave32 per (b,c) pair; coalesced stream of the 103MB scene tensor.
// ---------------------------------------------------------------------------
__global__ __launch_bounds__(256) void scene_reduce_kernel(
    const float* __restrict__ scene, const float* __restrict__ w_fc,
    float* __restrict__ q) {
  __shared__ float wfc[PP];
  int t = threadIdx.x;
  if (t < PP) wfc[t] = w_fc[t] * (1.0f / 16.0f);
  __syncthreads();
  int wave = t >> 5;
  int lane = t & 31;
  int pid  = blockIdx.x * 8 + wave;            // 0..8191 == b*256+c
  const float* base = scene + (size_t)pid * SHW;
  float acc = 0.f;
  for (int i = lane; i < SHW; i += 32) {
    int y = i / SH;
    int x = i - y * SH;
    acc += base[i] * wfc[(y >> 2) * 14 + (x >> 2)];
  }
  #pragma unroll
  for (int off = 16; off > 0; off >>= 1) acc += __shfl_xor(acc, off, 32);
  if (lane == 0) q[pid] = acc;
}

// ---------------------------------------------------------------------------
// Kernel B: v = w_scene*q + b_scene*sum(w_fc);  u = v^T*w_prop;
//           const[b] = b_prop . v[b] + b_fc.   (tiny; one block)
// ---------------------------------------------------------------------------
__global__ __launch_bounds__(256) void precompute_kernel(
    const float* __restrict__ q, const float* __restrict__ w_scene,
    const float* __restrict__ b_scene, const float* __restrict__ w_prop,
    const float* __restrict__ b_prop, const float* __restrict__ w_fc,
    const float* __restrict__ b_fc, float* __restrict__ u,
    float* __restrict__ constb) {
  __shared__ float v[B_SC * C_P];
  __shared__ float wsum_sh;
  int t = threadIdx.x;
  if (t == 0) {
    float s = 0.f;
    for (int p = 0; p < PP; ++p) s += w_fc[p];
    wsum_sh = s;
  }
  __syncthreads();
  float wsum = wsum_sh;
  for (int idx = t; idx < B_SC * C_P; idx += 256) {
    int b = idx >> 5, cp = idx & 31;
    float acc = b_scene[cp] * wsum;
    const float* wsrow = w_scene + cp * C_IN;
    const float* qb    = q + b * C_IN;
    for (int c = 0; c < C_IN; ++c) acc += wsrow[c] * qb[c];
    v[idx] = acc;
  }
  __syncthreads();
  for (int idx = t; idx < B_SC * C_IN; idx += 256) {
    int b = idx >> 8, c = idx & 255;
    float acc = 0.f;
    const float* vb = v + b * C_P;
    for (int cp = 0; cp < C_P; ++cp) acc += w_prop[cp * C_IN + c] * vb[cp];
    u[idx] = acc;
  }
  if (t < B_SC) {
    float acc = b_fc[0];
    const float* vb = v + t * C_P;
    for (int cp = 0; cp < C_P; ++cp) acc += b_prop[cp] * vb[cp];
    constb[t] = acc;
  }
}

// ---------------------------------------------------------------------------
// Kernel C: per proposal n:
//   logits[hw] = sum_c P[n,c,hw] * u[item_idx[n],c] + const[b]
//   out[n,c,hw] = (1 + sigmoid(logits[hw])) * P[n,c,hw]
// GEMV done with V_WMMA_F32_16X16X4_F32: 13 waves, one 16-spatial tile each.
// A layout (ISA 16x4 f32): lanes 0-15 K=0(v0)/1(v1), lanes 16-31 K=2/3.
// B columns all equal u[c0..c0+3] -> every D column = wanted dot product.
// D col 0: lane 0 holds M=0..7 (v0..v7), lane 16 holds M=8..15.
// ---------------------------------------------------------------------------
__global__ __launch_bounds__(416) void fused_attention_kernel(
    const float* __restrict__ proposal, const int* __restrict__ item_idx,
    const float* __restrict__ u, const float* __restrict__ constb,
    float* __restrict__ out) {
  __shared__ float u_sh[C_IN];
  __shared__ float logits[HWP];
  __shared__ float sig1[HW];
  __shared__ float cb_sh;
  int n = blockIdx.x;
  int t = threadIdx.x;
  const float* Pn = proposal + (size_t)n * (C_IN * HW);
  if (t < C_IN) {
    int b = item_idx[n];
    u_sh[t] = u[b * C_IN + t];
    if (t == 0) cb_sh = constb[b];
  }
  __syncthreads();

  // ---- Pass 1: logits via WMMA ----
  int wave   = t >> 5;        // 0..12 : spatial tile
  int lane   = t & 31;
  int laneLo = lane & 15;     // M within tile / N column
  int khalf  = lane >> 4;     // 0: K=0,1  1: K=2,3
  int col    = wave * 16 + laneLo;
  if (col > HW - 1) col = HW - 1;      // clamp tile 12 pad lanes in-bounds
  const float* p0 = Pn + col;

  v8f acc = {};
  for (int c0 = 0; c0 < C_IN; c0 += 4) {
    int r0 = c0 + 2 * khalf;
    v2f a, bf;
    a.x  = p0[(size_t)r0 * HW];
    a.y  = p0[(size_t)(r0 + 1) * HW];
    bf.x = u_sh[r0];
    bf.y = u_sh[r0 + 1];
    acc = __builtin_amdgcn_wmma_f32_16x16x4_f32(false, a, false, bf,
                                                (short)0, acc, false, false);
  }
  if (laneLo == 0) {
    int base = wave * 16 + khalf * 8;
    #pragma unroll
    for (int j = 0; j < 8; ++j) logits[base + j] = acc[j];
  }
  __syncthreads();
  if (t < HW) {
    float z = logits[t] + cb_sh;
    sig1[t] = 1.f + 1.f / (1.f + __expf(-z));
  }
  __syncthreads();

  // ---- Pass 2: residual scale, float4 vectorized (196 % 4 == 0) ----
  const f4* P4 = (const f4*)Pn;
  f4*       O4 = (f4*)(out + (size_t)n * (C_IN * HW));
  const int NV4 = C_IN * HW / 4;       // 12544
  for (int idx = t; idx < NV4; idx += 416) {
    int hw = (idx % 49) * 4;           // 49 float4 per 196-wide row
    f4 p = P4[idx];
    f4 o;
    o.x = p.x * sig1[hw + 0];
    o.y = p.y * sig1[hw + 1];
    o.z = p.z * sig1[hw + 2];
    o.w = p.w * sig1[hw + 3];
    O4[idx] = o;
  }
}

// ---------------------------------------------------------------------------
extern "C" void kernel_launch(void* const* d_in, const int* in_sizes, int n_in,
                              void* d_out, int out_size, void* d_ws,
                              size_t ws_size, hipStream_t stream) {
  const float* proposal = (const float*)d_in[0];
  const int*   item_idx = (const int*)d_in[1];
  const float* scene    = (const float*)d_in[2];
  const float* w_scene  = (const float*)d_in[3];
  const float* b_scene  = (const float*)d_in[4];
  const float* w_prop   = (const float*)d_in[5];
  const float* b_prop   = (const float*)d_in[6];
  const float* w_fc     = (const float*)d_in[7];
  const float* b_fc     = (const float*)d_in[8];
  float* out = (float*)d_out;

  float* q      = (float*)d_ws;            // 8192 floats
  float* u      = q + B_SC * C_IN;         // 8192 floats
  float* constb = u + B_SC * C_IN;         // 32 floats

  scene_reduce_kernel<<<dim3(1024), dim3(256), 0, stream>>>(scene, w_fc, q);
  precompute_kernel<<<dim3(1), dim3(256), 0, stream>>>(
      q, w_scene, b_scene, w_prop, b_prop, w_fc, b_fc, u, constb);
  fused_attention_kernel<<<dim3(N_PROP), dim3(416), 0, stream>>>(
      proposal, item_idx, u, constb, out);
}